// ChamferPccRateDistortionLoss_13194139533617
// MI455X (gfx1250) — compile-verified
//
#include <hip/hip_runtime.h>
#include <hip/hip_bf16.h>
#include <math.h>

typedef __attribute__((ext_vector_type(2))) float v2f;
typedef __attribute__((ext_vector_type(8))) float v8f;

// ---------------------------------------------------------------------------
// Phase 0: pack (c0, c1, c2) -> (c0, c1, c2, c0^2+c1^2+c2^2) as float4.
// This single format serves both WMMA operand roles:
//   A row (query):  (-2*c0, -2*c1, -2*c2, 1)      derived on the fly
//   B col (target): ( c0,    c1,    c2,  ||c||^2 ) used directly
// ---------------------------------------------------------------------------
__global__ __launch_bounds__(256) void pack_kernel(const float* __restrict__ src,
                                                   float4* __restrict__ dst, int n) {
    int i = blockIdx.x * blockDim.x + threadIdx.x;
    if (i < n) {
        float a = src[3 * i + 0];
        float b = src[3 * i + 1];
        float c = src[3 * i + 2];
        dst[i] = make_float4(a, b, c, a * a + b * b + c * c);
    }
}

// ---------------------------------------------------------------------------
// Phase 1: for each query point, min over all targets of squared L2 distance.
// One wave (32 lanes) handles a 16-row query tile and sweeps all Nt targets in
// 16-column tiles with V_WMMA_F32_16X16X4_F32:
//   C[m][n] = sum_k A[m][k]*B[k][n]  with A=(-2x,1), B=(y,||y||^2)
//           = ||y_n||^2 - 2 x_m . y_n
//   D[m][n] = C[m][n] + ||x_m||^2   (exact squared distance)
// Row-running-min kept in registers; reduced across lanes at the end.
//
// Operand layouts (ISA 7.12.2, f32, wave32):
//   A 16x4 : lane L<16 -> M=L, VGPR0=K0, VGPR1=K1 ; lane L>=16 -> M=L-16, K2/K3
//   B 4x16 : lane L    -> N=L&15 ; low half K0/K1, high half K2/K3
//   C 16x16: VGPR v, lanes 0-15 -> M=v, N=lane ; lanes 16-31 -> M=v+8, N=lane-16
// ---------------------------------------------------------------------------
__global__ __launch_bounds__(32) void nn_min_wmma(const float4* __restrict__ qp,
                                                  const float4* __restrict__ tp,
                                                  float* __restrict__ minout,
                                                  int Nq, int Nt) {
    const int lane = threadIdx.x;          // 0..31, full wave always active
    const int n    = lane & 15;            // column-within-tile / row-within-tile
    const int kp   = lane >> 4;            // which K pair this half-wave holds

    const int tile    = blockIdx.x;        // global 16-row tile index
    const int rowbase = tile * 16;         // global query row
    const int batch   = rowbase / Nq;      // tiles never straddle batches (Nq%16==0)
    const float4* tb  = tp + (size_t)batch * (size_t)Nt;

    // --- A operand (constant for the whole sweep) ---
    float4 qv = qp[rowbase + n];
    v2f a;
    if (kp == 0) { a.x = -2.0f * qv.x; a.y = -2.0f * qv.y; }
    else         { a.x = -2.0f * qv.z; a.y = 1.0f;         }

    // --- per-C-VGPR row norms ||x_m||^2 (row M = v + 8*kp for this lane) ---
    float x2r[8];
    const float* qf = (const float*)qp;
#pragma unroll
    for (int v = 0; v < 8; ++v)
        x2r[v] = qf[4 * (rowbase + v + 8 * kp) + 3];

    float rmin[8];
#pragma unroll
    for (int v = 0; v < 8; ++v) rmin[v] = 3.402823e38f;

    // --- sweep all targets, 16 columns per WMMA ---
#pragma unroll 4
    for (int j = 0; j < Nt; j += 16) {
        float4 tv = tb[j + n];             // (y0,y1,y2,||y||^2), L0/L2 resident
        v2f b;
        if (kp == 0) { b.x = tv.x; b.y = tv.y; }
        else         { b.x = tv.z; b.y = tv.w; }

        v8f c = {};
        c = __builtin_amdgcn_wmma_f32_16x16x4_f32(
                /*neg_a=*/false, a, /*neg_b=*/false, b,
                /*c_mod=*/(short)0, c, /*reuse_a=*/false, /*reuse_b=*/false);

#pragma unroll
        for (int v = 0; v < 8; ++v)
            rmin[v] = fminf(rmin[v], c[v] + x2r[v]);   // co-executes with WMMA
    }

    // --- reduce the 16-column partial mins across each 16-lane group ---
#pragma unroll
    for (int v = 0; v < 8; ++v) {
        float m = rmin[v];
        m = fminf(m, __shfl_xor(m, 8, 16));
        m = fminf(m, __shfl_xor(m, 4, 16));
        m = fminf(m, __shfl_xor(m, 2, 16));
        m = fminf(m, __shfl_xor(m, 1, 16));
        rmin[v] = m;
    }
    if (n == 0) {
#pragma unroll
        for (int v = 0; v < 8; ++v)
            minout[rowbase + v + 8 * kp] = rmin[v];    // lane0 -> M=v, lane16 -> M=v+8
    }
}

// ---------------------------------------------------------------------------
// Phase 2: reduce min arrays + log2(likelihoods) to the 3 scalar outputs.
// ---------------------------------------------------------------------------
__global__ __launch_bounds__(256) void finalize_kernel(const float* __restrict__ minx, int nx,
                                                       const float* __restrict__ miny, int ny,
                                                       const float* __restrict__ lik,  int nl,
                                                       float* __restrict__ out, float inv_np) {
    __shared__ float sx[256], sy[256], sl[256];
    const int t = threadIdx.x;
    float ax = 0.0f, ay = 0.0f, al = 0.0f;
    for (int i = t; i < nx; i += 256) ax += minx[i];
    for (int i = t; i < ny; i += 256) ay += miny[i];
    for (int i = t; i < nl; i += 256) al += log2f(lik[i]);
    sx[t] = ax; sy[t] = ay; sl[t] = al;
    __syncthreads();
    for (int s = 128; s > 0; s >>= 1) {
        if (t < s) { sx[t] += sx[t + s]; sy[t] += sy[t + s]; sl[t] += sl[t + s]; }
        __syncthreads();
    }
    if (t == 0) {
        float rec  = sx[0] / (float)nx + sy[0] / (float)ny;  // cham_x + cham_y
        float bpp  = -sl[0] * inv_np;                        // sum(log2)/(-num_points)
        float loss = bpp + 5.0f * rec;                       // LMBDA = 5
        out[0] = loss;
        out[1] = bpp;
        out[2] = rec;
    }
}

extern "C" void kernel_launch(void* const* d_in, const int* in_sizes, int n_in,
                              void* d_out, int out_size, void* d_ws, size_t ws_size,
                              hipStream_t stream) {
    const float* x_hat = (const float*)d_in[0];   // [B, M, 3]
    const float* points = (const float*)d_in[1];  // [B, P, 3]
    const float* lik = (const float*)d_in[2];     // [B, 1024]

    const int nX = in_sizes[0] / 3;   // B*M = 32768
    const int nP = in_sizes[1] / 3;   // B*P = 32768
    const int nL = in_sizes[2];       // B*1024 = 4096
    const int B  = nL / 1024;         // 4
    const int Mq = nX / B;            // 8192 queries per batch
    const int Mp = nP / B;            // 8192 targets per batch

    // Workspace layout (floats are 4B, float4 16B; hipMalloc gives 256B align):
    //   xpack : nX float4   (512 KB)
    //   ppack : nP float4   (512 KB)
    //   minx  : nX float    (128 KB)
    //   miny  : nP float    (128 KB)
    float4* xpack = (float4*)d_ws;
    float4* ppack = xpack + nX;
    float*  minx  = (float*)(ppack + nP);
    float*  miny  = minx + nX;

    pack_kernel<<<(nX + 255) / 256, 256, 0, stream>>>(x_hat, xpack, nX);
    pack_kernel<<<(nP + 255) / 256, 256, 0, stream>>>(points, ppack, nP);

    // Direction 1: each x_hat row finds its nearest point  -> cham_x
    nn_min_wmma<<<nX / 16, 32, 0, stream>>>(xpack, ppack, minx, Mq, Mp);
    // Direction 2: each point finds its nearest x_hat row  -> cham_y
    nn_min_wmma<<<nP / 16, 32, 0, stream>>>(ppack, xpack, miny, Mp, Mq);

    finalize_kernel<<<1, 256, 0, stream>>>(minx, nX, miny, nP, lik, nL,
                                           (float*)d_out, 1.0f / (float)nP);
}